// stepGenTUSmodel_42434276884815
// MI455X (gfx1250) — compile-verified
//
#include <hip/hip_runtime.h>
#include <cstdint>

// Problem constants (match reference)
#define B_ 8
#define S_ 512
#define V_ 32000
#define M_ 32
#define PAD_ID_ 1

// Parallel decomposition: grid (B_, CHUNKS_) blocks of 1024 threads (32 waves).
// Each wave handles ROWS_PER_WAVE_ = 2 rows -> 2048 waves, ~2 latency chains each.
#define CHUNKS_ 8
#define ROWS_PER_BLOCK_ (S_ / CHUNKS_)                    // 64
#define WAVES_PER_BLOCK_ 32
#define ROWS_PER_WAVE_ (ROWS_PER_BLOCK_ / WAVES_PER_BLOCK_) // 2

// logp[b,s] = l_a - m - log( sum_{mask>0} exp(l_mask - m) )   (softmax Z cancels)
__global__ __launch_bounds__(1024, 1)
void logp_gather_kernel(const float* __restrict__ logits,      // [B,S,V] f32
                        const long long* __restrict__ a,       // [B,S]   i64
                        const long long* __restrict__ mask,    // [B,S,M] i64
                        float* __restrict__ partial)           // [B_*CHUNKS_] f32
{
    // Double-buffered per-wave staging of mask rows via the async copy engine.
    __shared__ long long idxbuf[WAVES_PER_BLOCK_][ROWS_PER_WAVE_][M_]; // 16 KB
    __shared__ float wave_part[WAVES_PER_BLOCK_];

    const int b     = blockIdx.x;
    const int chunk = blockIdx.y;
    const int tid   = (int)threadIdx.x;
    const int wave  = tid >> 5;
    const int lane  = tid & 31;

    const int s0 = chunk * ROWS_PER_BLOCK_ + wave * ROWS_PER_WAVE_;

    // Issue ROWS_PER_WAVE_ async global->LDS copies (8 B per lane, 256 B per row).
    // GV mode: per-lane 64-bit global address, LDS dest byte-address in first VGPR.
    #pragma unroll
    for (int it = 0; it < ROWS_PER_WAVE_; ++it) {
        const long long* gsrc = mask + ((size_t)(b * S_ + (s0 + it)) * M_) + lane;
        unsigned lds_addr = (unsigned)(uintptr_t)(&idxbuf[wave][it][lane]); // low 32b = LDS offset
        asm volatile("global_load_async_to_lds_b64 %0, %1, off"
                     :
                     : "v"(lds_addr), "v"((unsigned long long)(uintptr_t)gsrc)
                     : "memory");
    }

    float acc = 0.0f;
    #pragma unroll
    for (int it = 0; it < ROWS_PER_WAVE_; ++it) {
        // Async loads complete in order: asynccnt<=1 releases buffer 0 while
        // buffer 1 is still in flight (software pipelining on ASYNCcnt).
        if (it == 0) { asm volatile("s_wait_asynccnt 1" ::: "memory"); }
        else         { asm volatile("s_wait_asynccnt 0" ::: "memory"); }

        const int s = s0 + it;
        const float* row = logits + (size_t)(b * S_ + s) * V_;

        const long long id = idxbuf[wave][it][lane];   // lane j <-> candidate j (M_ == wave32)
        const bool valid = (id > 0);
        float l = valid ? row[id] : -INFINITY;

        // Wave-wide max (butterfly, all lanes end with m)
        float m = l;
        #pragma unroll
        for (int o = 16; o > 0; o >>= 1) m = fmaxf(m, __shfl_xor(m, o));

        // Wave-wide sum of exp(l - m) over valid lanes
        float e = valid ? expf(l - m) : 0.0f;
        #pragma unroll
        for (int o = 16; o > 0; o >>= 1) e += __shfl_xor(e, o);

        const long long av = a[b * S_ + s];            // wave-uniform
        const float la = row[av];

        const float logp = la - m - logf(e);
        const float padf = (av != PAD_ID_) ? 1.0f : 0.0f; // 0*inf -> nan, matches ref
        acc += logp * padf;
    }

    // acc is wave-uniform after the butterflies; deterministic block reduction.
    if (lane == 0) wave_part[wave] = acc;
    __syncthreads();
    if (wave == 0) {
        float v = wave_part[lane];
        #pragma unroll
        for (int o = 16; o > 0; o >>= 1) v += __shfl_xor(v, o);
        if (lane == 0) partial[b * CHUNKS_ + chunk] = v;
    }
}

// Fixed-order final sum -> bitwise-deterministic across graph replays.
__global__ void logp_finalize_kernel(const float* __restrict__ partial,
                                     float* __restrict__ out)
{
    const int b = (int)threadIdx.x;
    if (b < B_) {
        float s = 0.0f;
        #pragma unroll
        for (int c = 0; c < CHUNKS_; ++c) s += partial[b * CHUNKS_ + c];
        out[b] = s;
    }
}

extern "C" void kernel_launch(void* const* d_in, const int* in_sizes, int n_in,
                              void* d_out, int out_size, void* d_ws, size_t ws_size,
                              hipStream_t stream)
{
    (void)in_sizes; (void)n_in; (void)out_size; (void)ws_size;

    const float*     logits = (const float*)d_in[0];     // [B,S,V] float32
    const long long* a      = (const long long*)d_in[1]; // [B,S]   int64
    const long long* mask   = (const long long*)d_in[2]; // [B,S,M] int64
    float* out     = (float*)d_out;                      // [B] float32
    float* partial = (float*)d_ws;                       // B_*CHUNKS_ floats scratch

    dim3 grid(B_, CHUNKS_);
    logp_gather_kernel<<<grid, 1024, 0, stream>>>(logits, a, mask, partial);
    logp_finalize_kernel<<<1, 32, 0, stream>>>(partial, out);
}